// OccupancyGrid_41188736368829
// MI455X (gfx1250) — compile-verified
//
#include <hip/hip_runtime.h>
#include <hip/hip_bf16.h>
#include <stdint.h>

#define GRID_N      128
#define MASK_DWORDS (GRID_N * GRID_N * GRID_N / 32)   // 65536 dwords = 256 KiB
#define MASK_UINT4  (MASK_DWORDS / 4)                 // 16384 x 16B
#define LDS_BYTES   (MASK_DWORDS * 4)                 // 262144 B of the 320 KiB/WGP
#define SAMPLE_BLOCKS  512
#define SAMPLE_THREADS 1024

typedef int v4i __attribute__((vector_size(16)));
typedef __attribute__((address_space(1))) v4i* gptr_v4i;
typedef __attribute__((address_space(3))) v4i* lptr_v4i;

// ---------------------------------------------------------------------------
// Kernel 1: pack the binary float grid (0.0 / 1.0) into a 1-bit/voxel mask.
// Wave32 ballot: lane L of each wave contributes bit L of one mask dword.
// ---------------------------------------------------------------------------
__global__ void occ_pack_kernel(const float* __restrict__ grid,
                                uint32_t* __restrict__ mask, int n)
{
    int i = blockIdx.x * blockDim.x + threadIdx.x;
    bool occ = (i < n) && (grid[i] != 0.0f);
    uint32_t m = (uint32_t)__ballot(occ);          // wave32: low 32 bits valid
    if (((threadIdx.x & 31) == 0) && (i < n))
        mask[i >> 5] = m;
}

// ---------------------------------------------------------------------------
// Kernel 2: stage the 256 KiB bitmask into LDS via CDNA5 async global->LDS
// DMA, then trilinear-sample every point out of LDS bit tests.
// ---------------------------------------------------------------------------
__global__ void __launch_bounds__(SAMPLE_THREADS, 1)
occ_sample_kernel(const float* __restrict__ coords,
                  const uint32_t* __restrict__ gmask,
                  float* __restrict__ out, int npts)
{
    extern __shared__ uint32_t smask[];            // 65536 dwords
    const int tid = threadIdx.x;

    // ---- cooperative 256 KiB global -> LDS staging -------------------------
#if __has_builtin(__builtin_amdgcn_global_load_async_to_lds_b128)
    {
        gptr_v4i gsrc = (gptr_v4i)gmask;           // drops const; AS1 int4*
        lptr_v4i ldst = (lptr_v4i)smask;           // AS3 int4*
        for (int k = tid; k < MASK_UINT4; k += SAMPLE_THREADS) {
            __builtin_amdgcn_global_load_async_to_lds_b128(
                gsrc + k, ldst + k, /*offset=*/0, /*cpol=*/0);
        }
#if __has_builtin(__builtin_amdgcn_s_wait_asynccnt)
        __builtin_amdgcn_s_wait_asynccnt(0);
#else
        asm volatile("s_wait_asynccnt 0" ::: "memory");
#endif
    }
#else
    {
        const uint4* g4 = (const uint4*)gmask;
        uint4* s4 = (uint4*)smask;
        for (int k = tid; k < MASK_UINT4; k += SAMPLE_THREADS)
            s4[k] = g4[k];
    }
#endif
    __syncthreads();

    // ---- trilinear sampling out of the LDS bitmask -------------------------
    auto bitat = [&](int zz, int yy, int xx) -> float {
        if ((unsigned)xx < (unsigned)GRID_N &&
            (unsigned)yy < (unsigned)GRID_N &&
            (unsigned)zz < (unsigned)GRID_N) {
            int idx = (zz << 14) | (yy << 7) | xx;   // z*128*128 + y*128 + x
            return (float)((smask[idx >> 5] >> (idx & 31)) & 1u);
        }
        return 0.0f;                                  // zeros padding
    };

    const int stride = gridDim.x * blockDim.x;
    for (int i = blockIdx.x * blockDim.x + tid; i < npts; i += stride) {
        float x = coords[3 * i + 0];
        float y = coords[3 * i + 1];
        float z = coords[3 * i + 2];

        // ((c + 1) * N - 1) * 0.5, align_corners=False  (N == 128 all dims)
        float ix = ((x + 1.0f) * 128.0f - 1.0f) * 0.5f;
        float iy = ((y + 1.0f) * 128.0f - 1.0f) * 0.5f;
        float iz = ((z + 1.0f) * 128.0f - 1.0f) * 0.5f;

        float fx = floorf(ix), fy = floorf(iy), fz = floorf(iz);
        float wx = ix - fx,    wy = iy - fy,    wz = iz - fz;
        int x0 = (int)fx, y0 = (int)fy, z0 = (int)fz;

        float owx = 1.0f - wx, owy = 1.0f - wy, owz = 1.0f - wz;

        float r =
            bitat(z0,     y0,     x0    ) * owz * owy * owx +
            bitat(z0,     y0,     x0 + 1) * owz * owy * wx  +
            bitat(z0,     y0 + 1, x0    ) * owz * wy  * owx +
            bitat(z0,     y0 + 1, x0 + 1) * owz * wy  * wx  +
            bitat(z0 + 1, y0,     x0    ) * wz  * owy * owx +
            bitat(z0 + 1, y0,     x0 + 1) * wz  * owy * wx  +
            bitat(z0 + 1, y0 + 1, x0    ) * wz  * wy  * owx +
            bitat(z0 + 1, y0 + 1, x0 + 1) * wz  * wy  * wx;

        out[i] = r;
    }
}

// ---------------------------------------------------------------------------
extern "C" void kernel_launch(void* const* d_in, const int* in_sizes, int n_in,
                              void* d_out, int out_size, void* d_ws, size_t ws_size,
                              hipStream_t stream)
{
    const float* grid   = (const float*)d_in[0];   // [128,128,128] f32 (binary)
    const float* coords = (const float*)d_in[1];   // [N,3] f32
    float* out = (float*)d_out;                    // [N] f32
    uint32_t* mask = (uint32_t*)d_ws;              // 256 KiB bitmask scratch

    const int ngrid = in_sizes[0];                 // 2,097,152 voxels
    const int npts  = out_size;                    // 8,388,608 points

    (void)hipFuncSetAttribute(
        reinterpret_cast<const void*>(occ_sample_kernel),
        hipFuncAttributeMaxDynamicSharedMemorySize, LDS_BYTES);

    const int packThreads = 256;
    const int packBlocks  = (ngrid + packThreads - 1) / packThreads;
    occ_pack_kernel<<<packBlocks, packThreads, 0, stream>>>(grid, mask, ngrid);

    occ_sample_kernel<<<SAMPLE_BLOCKS, SAMPLE_THREADS, LDS_BYTES, stream>>>(
        coords, mask, out, npts);
}